// DiffMPC2_28424093565700
// MI455X (gfx1250) — compile-verified
//
#include <hip/hip_runtime.h>

#ifndef __has_builtin
#define __has_builtin(x) 0
#endif

#define MPC_LR 0.01f

// Pointer types matching the gfx1250 async LDS-DMA builtin signatures
// (both global-side parameters are non-const v4i AS(1)*):
//   global_load_async_to_lds_b128(v4i AS(1)*, v4i AS(3)*, Imm, Imm)
//   global_store_async_from_lds_b128(v4i AS(1)*, v4i AS(3)*, Imm, Imm)
typedef int v4i __attribute__((ext_vector_type(4)));
typedef __attribute__((address_space(1))) v4i gbl_v4i;
typedef __attribute__((address_space(3))) v4i lds_v4i;

#if defined(__gfx1250__) &&                                             \
    __has_builtin(__builtin_amdgcn_global_load_async_to_lds_b128) &&    \
    __has_builtin(__builtin_amdgcn_global_store_async_from_lds_b128)
#define MPC_ASYNC_LDS 1
#endif

// One row (4 floats) per lane. u100 = A*u0 + G*b with
//   a = 1 - 2*LR*q,  b = -LR*p,  A = a^100,  G = sum_{k<100} a^k
// built by exponent doubling: A2n = An^2, G2n = Gn*(1+An); 100 = 64+32+4.
__global__ __launch_bounds__(256) void diffmpc_u100(
    const float* __restrict__ Q,
    const float* __restrict__ P,
    const float* __restrict__ Uin,
    float*       __restrict__ Uout,
    int nRows, int qpStride4, int qpOff4)
{
#ifdef MPC_ASYNC_LDS
  __shared__ float4 lds_u[256];
  __shared__ float4 lds_o[256];
#endif
  const int tid = threadIdx.x;
  const int row = blockIdx.x * 256 + tid;
  if (row >= nRows) return;

  const float4* __restrict__ q4 = reinterpret_cast<const float4*>(Q);
  const float4* __restrict__ p4 = reinterpret_cast<const float4*>(P);
  const float4* __restrict__ u4 = reinterpret_cast<const float4*>(Uin);
  float4*       __restrict__ o4 = reinterpret_cast<float4*>(Uout);

  const int qpIdx = row * qpStride4 + qpOff4;   // last float4 of the row = q_u / p_u

  // Warm the cache hierarchy for the strided q/p rows (global_prefetch_b8).
  __builtin_prefetch(q4 + qpIdx, 0, 3);
  __builtin_prefetch(p4 + qpIdx, 0, 3);

#ifdef MPC_ASYNC_LDS
  // Async DMA of this lane's u_init row into LDS (tracked by ASYNCcnt),
  // overlapped with the q/p vector loads below.
  __builtin_amdgcn_global_load_async_to_lds_b128(
      (gbl_v4i*)(u4 + row), (lds_v4i*)&lds_u[tid], 0, 0);
#endif

  const float4 qv = q4[qpIdx];
  const float4 pv = p4[qpIdx];

  float4 uv;
#ifdef MPC_ASYNC_LDS
  #if __has_builtin(__builtin_amdgcn_s_wait_asynccnt)
    __builtin_amdgcn_s_wait_asynccnt(0);
  #else
    asm volatile("s_wait_asynccnt 0" ::: "memory");
  #endif
  uv = lds_u[tid];
#else
  uv = u4[row];
#endif

  float a[4]  = { fmaf(-2.0f * MPC_LR, qv.x, 1.0f),
                  fmaf(-2.0f * MPC_LR, qv.y, 1.0f),
                  fmaf(-2.0f * MPC_LR, qv.z, 1.0f),
                  fmaf(-2.0f * MPC_LR, qv.w, 1.0f) };
  float b[4]  = { -MPC_LR * pv.x, -MPC_LR * pv.y,
                  -MPC_LR * pv.z, -MPC_LR * pv.w };
  float u0[4] = { uv.x, uv.y, uv.z, uv.w };

  float A[4], G[4], A4s[4], G4s[4], A32s[4], G32s[4];
  #pragma unroll
  for (int c = 0; c < 4; ++c) { A[c] = a[c]; G[c] = 1.0f; }

  // e = 1 -> 2 -> 4
  #pragma unroll
  for (int d = 0; d < 2; ++d)
    #pragma unroll
    for (int c = 0; c < 4; ++c) { G[c] = fmaf(G[c], A[c], G[c]); A[c] *= A[c]; }
  #pragma unroll
  for (int c = 0; c < 4; ++c) { A4s[c] = A[c]; G4s[c] = G[c]; }

  // e = 4 -> 8 -> 16 -> 32
  #pragma unroll
  for (int d = 0; d < 3; ++d)
    #pragma unroll
    for (int c = 0; c < 4; ++c) { G[c] = fmaf(G[c], A[c], G[c]); A[c] *= A[c]; }
  #pragma unroll
  for (int c = 0; c < 4; ++c) { A32s[c] = A[c]; G32s[c] = G[c]; }

  // e = 32 -> 64
  #pragma unroll
  for (int c = 0; c < 4; ++c) { G[c] = fmaf(G[c], A[c], G[c]); A[c] *= A[c]; }

  // 64 + 32 -> 96 : G96 = G64 + A64*G32 ; A96 = A64*A32
  #pragma unroll
  for (int c = 0; c < 4; ++c) { G[c] = fmaf(A[c], G32s[c], G[c]); A[c] *= A32s[c]; }
  // 96 + 4 -> 100
  #pragma unroll
  for (int c = 0; c < 4; ++c) { G[c] = fmaf(A[c], G4s[c], G[c]); A[c] *= A4s[c]; }

  float r[4];
  #pragma unroll
  for (int c = 0; c < 4; ++c) r[c] = fmaf(A[c], u0[c], G[c] * b[c]);

#ifdef MPC_ASYNC_LDS
  lds_o[tid] = make_float4(r[0], r[1], r[2], r[3]);
  // The async engine is unordered with this wave's DS store: drain DScnt first.
  #if __has_builtin(__builtin_amdgcn_s_wait_dscnt)
    __builtin_amdgcn_s_wait_dscnt(0);
  #else
    asm volatile("s_wait_dscnt 0" ::: "memory");
  #endif
  __builtin_amdgcn_global_store_async_from_lds_b128(
      (gbl_v4i*)(o4 + row), (lds_v4i*)&lds_o[tid], 0, 0);
  // Completion covered by the implicit wait-idle at s_endpgm.
#else
  o4[row] = make_float4(r[0], r[1], r[2], r[3]);
#endif
}

extern "C" void kernel_launch(void* const* d_in, const int* in_sizes, int n_in,
                              void* d_out, int out_size, void* d_ws, size_t ws_size,
                              hipStream_t stream) {
  (void)d_ws; (void)ws_size; (void)n_in;
  // setup_inputs order: x_init (B,S) [unused], Q (B,S+C), p (B,S+C), u_init (B,C)
  const float* Q = (const float*)d_in[1];
  const float* P = (const float*)d_in[2];
  const float* U = (const float*)d_in[3];
  float* out = (float*)d_out;

  const int C = 4;
  const int B = out_size / C;                            // 1048576
  const int SC = (B > 0) ? (in_sizes[1] / B) : 16;       // S + C = 16
  const int stride4 = SC / 4;                            // row stride in float4
  const int off4    = stride4 - 1;                       // q_u/p_u = last float4

  dim3 block(256);
  dim3 grid((B + 255) / 256);
  diffmpc_u100<<<grid, block, 0, stream>>>(Q, P, U, out, B, stride4, off4);
}